// LatentCapsLayer_56298431316284
// MI455X (gfx1250) — compile-verified
//
#include <hip/hip_runtime.h>

// ---------------------------------------------------------------------------
// EM-routing capsule layer for MI455X (gfx1250).
// Votes are recomputed per pass with V_WMMA_F32_16X16X4_F32 (fp32 WMMA):
// per (b,i) block it is a 16(M=xy) x 1024(N=c*64+l) x 16(K=p) GEMM done as
// 64 tiles of 16x16, each = 4 chained K=4 WMMAs. W (64MB) is repacked once
// into Wt[i][col][p] (p innermost -> paired b64 B-frag loads, one 64B line
// per lane-column) and stays resident in the 192MB L2, so recomputation is
// ~free in HBM BW versus materializing the 268MB vote tensor. A-fragments
// are hoisted into registers (tile-invariant); per tile the B-fragments are
// issued as one 4 x global_load_b64 clause before the WMMA chain so their
// latencies overlap, then 4 x v_wmma + paired D-tile stores.
// ---------------------------------------------------------------------------

typedef float vf2 __attribute__((ext_vector_type(2)));
typedef float vf8 __attribute__((ext_vector_type(8)));

#define B_CAPS   1024
#define NTOT     16384          // B_CAPS * 16 positions
#define XBATCH   278528         // 17408 * 16 floats per batch of x
#define ACT_OFF  262144         // offset of activations inside one batch of x
#define CL       1024           // C_CAPS * L_VEC
#define LOG_2PI  1.8378770664093453f

// workspace layout (float offsets)
#define OFF_R      0u                         // 1,048,576  : R[b][n][c]
#define OFF_WT     1048576u                   // 16,777,216 : Wt[i][c*64+l][p]
#define OFF_STATS  (OFF_WT + 16777216u)       // sumR[64] | sumRV[4096] | sumRV2[4096]
#define STATS_N    8256u
#define OFF_MU     (OFF_STATS + STATS_N)      // 4096 : mu[b][c][l]
#define OFF_LOGAC  (OFF_MU + 4096u)           // 64   : log a_c
#define OFF_CONST  (OFF_LOGAC + 64u)          // 64   : 0.5*sum_l(log2pi+log sigma)

// ---------------------------------------------------------------------------
__global__ void caps_init_r(float* __restrict__ R) {
  R[blockIdx.x * 256u + threadIdx.x] = 0.0625f;   // 1/C
}

__global__ void caps_zero(float* __restrict__ p, unsigned n) {
  const unsigned idx = blockIdx.x * 256u + threadIdx.x;
  if (idx < n) p[idx] = 0.f;
}

// W[c][i][l][p] -> Wt[i][c*64+l][p]  (p innermost on BOTH sides: repack is
// fully coalesced, and B-frag loads become contiguous b64 pairs).
__global__ __launch_bounds__(256) void caps_repack_w(const float* __restrict__ W,
                                                     float* __restrict__ Wt) {
  const unsigned idx = blockIdx.x * 256u + threadIdx.x;   // 16,777,216 total
  const unsigned i   = idx >> 14;
  const unsigned col = (idx >> 4) & 1023u;   // col = c*64 + l
  const unsigned p   = idx & 15u;
  const unsigned c = col >> 6, l = col & 63u;
  Wt[idx] = W[((c * 1024u + i) * 64u + l) * 16u + p];
}

// ---------------------------------------------------------------------------
// WMMA vote-tile computation for one (b,i) block, one 512-column half.
// af[kg]: tile-invariant A fragments (hoisted by the caller).
// V_s: [xy*512 + colLocal]
__device__ __forceinline__ void compute_votes_half(
    const vf2 af[4], const float* __restrict__ Wt,
    float* __restrict__ V_s, int i, int half, int wv, int lane) {
  const int m  = lane & 15;
  const int hi = (lane >> 4) & 1;
  #pragma unroll
  for (int tt = 0; tt < 4; ++tt) {
    const int tl = wv * 4 + tt;          // local tile 0..31
    const int t  = half * 32 + tl;       // global tile 0..63
    // lane's B column: 16 consecutive floats (one 64B line) hold all p
    const float* wbase =
        Wt + (((unsigned)i * 1024u + (unsigned)(t * 16 + m)) << 4) + hi * 2;
    // Issue all four B-frag loads up front (one clause, overlapped latency).
    vf2 bfrag[4];
    #pragma unroll
    for (int kg = 0; kg < 4; ++kg)
      bfrag[kg] = *reinterpret_cast<const vf2*>(wbase + kg * 4);
    // Chained WMMAs: B 4x16 f32: v0 = K={0|2}, v1 = K={1|3}, N = lane&15
    vf8 acc = {};
    #pragma unroll
    for (int kg = 0; kg < 4; ++kg)
      acc = __builtin_amdgcn_wmma_f32_16x16x4_f32(
          false, af[kg], false, bfrag[kg], (short)0, acc, false, false);
    // D 16x16 f32: VGPR r -> M = r + 8*hi, N = lane&15
    const int colL = tl * 16 + m;
    const int row0 = hi ? 8 : 0;
    #pragma unroll
    for (int r = 0; r < 8; ++r) V_s[(row0 + r) * 512 + colL] = acc[r];
  }
}

// Load the 4 tile-invariant A fragments from pose_s[p*16 + xy].
__device__ __forceinline__ void load_afrags(const float* __restrict__ pose_s,
                                            vf2 af[4], int lane) {
  const int m  = lane & 15;
  const int hi = (lane >> 4) & 1;
  #pragma unroll
  for (int kg = 0; kg < 4; ++kg) {
    const int pa = kg * 4 + hi * 2;
    af[kg].x = pose_s[pa * 16 + m];        // A 16x4 f32: v0 = K={0|2}
    af[kg].y = pose_s[(pa + 1) * 16 + m];  //             v1 = K={1|3}
  }
}

// ---------------------------------------------------------------------------
// Stats pass: accumulate sumR[b,c], sumRV[b,c,l], sumRV2[b,c,l] via atomics.
__global__ __launch_bounds__(256) void caps_stats(
    const float* __restrict__ x, const float* __restrict__ Wt,
    const float* __restrict__ R, float* __restrict__ sumR,
    float* __restrict__ sumRV, float* __restrict__ sumRV2) {
  __shared__ float pose_s[256];
  __shared__ float r_s[256];           // r_hat[xy][c]
  __shared__ float V_s[16 * 512];

  const int i = blockIdx.x, b = blockIdx.y;
  const int tid = threadIdx.x;
  const int lane = tid & 31, wv = tid >> 5;

  pose_s[tid] = x[b * XBATCH + ((tid >> 4) * B_CAPS + i) * 16 + (tid & 15)];
  {
    const int xy = tid >> 4, c = tid & 15;
    const int n = i * 16 + xy;
    const float a = x[b * XBATCH + ACT_OFF + n];
    const float r = R[((unsigned)b * NTOT + n) * 16u + c];
    r_s[xy * 16 + c] = fmaxf(r * a, 0.01f);
  }
  __syncthreads();

  float sR = 0.f;
  if (tid < 16) {
    #pragma unroll
    for (int mm = 0; mm < 16; ++mm) sR += r_s[mm * 16 + tid];
  }

  vf2 af[4];
  load_afrags(pose_s, af, lane);

  for (int half = 0; half < 2; ++half) {
    compute_votes_half(af, Wt, V_s, i, half, wv, lane);
    __syncthreads();
    #pragma unroll
    for (int q = 0; q < 2; ++q) {
      const int colL = tid + q * 256;
      const int col = half * 512 + colL;
      const int c = col >> 6;
      float s1 = 0.f, s2 = 0.f;
      #pragma unroll
      for (int mm = 0; mm < 16; ++mm) {
        const float v = V_s[mm * 512 + colL];
        const float r = r_s[mm * 16 + c];
        s1 = fmaf(r, v, s1);
        s2 = fmaf(r * v, v, s2);
      }
      atomicAdd(&sumRV[(b * 16 + c) * 64 + (col & 63)], s1);
      atomicAdd(&sumRV2[(b * 16 + c) * 64 + (col & 63)], s2);
    }
    __syncthreads();
  }
  if (tid < 16) atomicAdd(&sumR[b * 16 + tid], sR);
}

// ---------------------------------------------------------------------------
// Finalize: mu, sigma (E[V^2]-mu^2), cost, a_c; optionally write output.
__global__ void caps_finalize(
    const float* __restrict__ sumR, const float* __restrict__ sumRV,
    const float* __restrict__ sumRV2, const float* __restrict__ beta_v,
    const float* __restrict__ beta_a, const float* __restrict__ lambda_,
    float* __restrict__ mu, float* __restrict__ logac,
    float* __restrict__ cconst, float* __restrict__ out, int write_out) {
  const int tid = threadIdx.x;         // 64 threads = (b,c)
  if (tid >= 64) return;
  const int b = tid >> 4, c = tid & 15;
  const float bv = beta_v[0], lam = lambda_[0];
  const float sr = sumR[tid];          // >= 163.84, never zero
  float cost = 0.f, cns = 0.f;
  for (int l = 0; l < 64; ++l) {
    const float muv = sumRV[tid * 64 + l] / sr;
    const float ex2 = sumRV2[tid * 64 + l] / sr;
    const float sg = fmaxf(ex2 - muv * muv, 0.01f);
    mu[tid * 64 + l] = muv;
    const float lg = logf(sg);
    cost += bv + lg;
    cns += 0.5f * (LOG_2PI + lg);
  }
  cost *= sr;
  const float ac = 1.f / (1.f + expf(-lam * (beta_a[c] - cost)));
  logac[tid] = logf(ac);
  cconst[tid] = cns;
  if (write_out) {
    for (int l = 0; l < 64; ++l) out[b * 1040 + c * 64 + l] = mu[tid * 64 + l];
    out[b * 1040 + 1024 + c] = ac;
  }
}

// ---------------------------------------------------------------------------
// R update: recompute votes (WMMA), logp = -sum_l (V-mu)^2 - const_c,
// R = softmax_c(log a_c + logp).
__global__ __launch_bounds__(256) void caps_rupdate(
    const float* __restrict__ x, const float* __restrict__ Wt,
    const float* __restrict__ mu, const float* __restrict__ logac,
    const float* __restrict__ cconst, float* __restrict__ R) {
  __shared__ float pose_s[256];
  __shared__ float mu_s[1024];
  __shared__ float logit_s[256];
  __shared__ float V_s[16 * 512];

  const int i = blockIdx.x, b = blockIdx.y;
  const int tid = threadIdx.x;
  const int lane = tid & 31, wv = tid >> 5;
  const int xy = tid >> 4, c = tid & 15;

  pose_s[tid] = x[b * XBATCH + ((tid >> 4) * B_CAPS + i) * 16 + (tid & 15)];
  #pragma unroll
  for (int q = 0; q < 4; ++q) mu_s[tid + q * 256] = mu[b * CL + tid + q * 256];
  __syncthreads();

  const float lac = logac[b * 16 + c];
  const float cc = cconst[b * 16 + c];

  vf2 af[4];
  load_afrags(pose_s, af, lane);

  for (int half = 0; half < 2; ++half) {
    compute_votes_half(af, Wt, V_s, i, half, wv, lane);
    __syncthreads();
    if ((c >> 3) == half) {            // this half's columns cover 8 classes
      const float* vrow = &V_s[xy * 512 + (c & 7) * 64];
      const float* mrow = &mu_s[c * 64];
      float s = 0.f;
      #pragma unroll
      for (int l = 0; l < 64; ++l) {
        const float d = vrow[l] - mrow[l];
        s = fmaf(-d, d, s);
      }
      logit_s[xy * 16 + c] = lac + s - cc;
    }
    __syncthreads();
  }

  float mx = -3.4e38f;
  #pragma unroll
  for (int k = 0; k < 16; ++k) mx = fmaxf(mx, logit_s[xy * 16 + k]);
  float den = 0.f;
  #pragma unroll
  for (int k = 0; k < 16; ++k) den += expf(logit_s[xy * 16 + k] - mx);
  const float rnew = expf(logit_s[xy * 16 + c] - mx) / den;
  R[((unsigned)b * NTOT + i * 16 + xy) * 16u + c] = rnew;
}

// ---------------------------------------------------------------------------
extern "C" void kernel_launch(void* const* d_in, const int* in_sizes, int n_in,
                              void* d_out, int out_size, void* d_ws, size_t ws_size,
                              hipStream_t stream) {
  const float* x       = (const float*)d_in[0];
  const float* W       = (const float*)d_in[1];
  const float* beta_v  = (const float*)d_in[2];
  const float* beta_a  = (const float*)d_in[3];
  const float* lambda_ = (const float*)d_in[4];
  float* out = (float*)d_out;
  float* ws  = (float*)d_ws;

  float* R      = ws + OFF_R;
  float* Wt     = ws + OFF_WT;
  float* sumR   = ws + OFF_STATS;
  float* sumRV  = sumR + 64;
  float* sumRV2 = sumRV + 4096;
  float* mu     = ws + OFF_MU;
  float* logac  = ws + OFF_LOGAC;
  float* cconst = ws + OFF_CONST;

  caps_init_r<<<4096, 256, 0, stream>>>(R);
  caps_repack_w<<<65536, 256, 0, stream>>>(W, Wt);

  const dim3 grid(B_CAPS, 4);   // (i, b)
  for (int iter = 0; iter < 2; ++iter) {
    caps_zero<<<(STATS_N + 255) / 256, 256, 0, stream>>>(sumR, STATS_N);
    caps_stats<<<grid, 256, 0, stream>>>(x, Wt, R, sumR, sumRV, sumRV2);
    caps_finalize<<<1, 64, 0, stream>>>(sumR, sumRV, sumRV2, beta_v, beta_a,
                                        lambda_, mu, logac, cconst, out,
                                        iter == 1 ? 1 : 0);
    if (iter == 0)
      caps_rupdate<<<grid, 256, 0, stream>>>(x, Wt, mu, logac, cconst, R);
  }
}